// TestSheafConv_89850715832320
// MI455X (gfx1250) — compile-verified
//
#include <hip/hip_runtime.h>
#include <hip/hip_bf16.h>
#include <math.h>

#define HIDDEN 64
#define DIMF   2
#define NLAYER 3
#define NNODE  50000
#define NEDGE  800000
#define NGRAPH 1024
#define CDIM   128            // HIDDEN*DIMF
#define M2     (NNODE * DIMF) // 100000 rows of the h-GEMM

typedef float v2f __attribute__((ext_vector_type(2)));
typedef float v8f __attribute__((ext_vector_type(8)));
typedef unsigned int u32x4 __attribute__((ext_vector_type(4)));
typedef int i32x4 __attribute__((ext_vector_type(4)));
typedef int i32x8 __attribute__((ext_vector_type(8)));

__device__ __forceinline__ void atomic_add_f32(float* p, float v) {
  __hip_atomic_fetch_add(p, v, __ATOMIC_RELAXED, __HIP_MEMORY_SCOPE_AGENT);
}

// ---------------------------------------------------------------------------
// Tensor Data Mover: DMA a 64x64 f32 tile (row-major, stride 64) into LDS.
// D# layout per CDNA5 ISA ch.8 (group0: count/lds/global/type, group1: dims).
#if __has_builtin(__builtin_amdgcn_tensor_load_to_lds)
#define HAVE_TDM 1
__device__ __forceinline__ void tdm_load_w64x64(const float* gW, void* lds_ptr) {
  unsigned lds_addr = (unsigned)(size_t)lds_ptr;  // flat LDS addr low 32b = offset
  unsigned long long ga = (unsigned long long)(size_t)gW;
  u32x4 g0;
  g0[0] = 1u;                                        // count=1, user descriptor
  g0[1] = lds_addr;                                  // lds_addr [63:32]
  g0[2] = (unsigned)(ga & 0xffffffffu);              // global_addr [95:64]
  g0[3] = (unsigned)((ga >> 32) & 0x1ffffffu)        // global_addr [120:96]
        | (2u << 30);                                // type=2 ("image")
  i32x8 g1;
  g1[0] = (int)(2u << 16);   // workgroup_mask=0, data_size=2 (4B), no flags
  g1[1] = (int)(64u << 16);  // atomic_barrier_addr=0 | tensor_dim0 lo16 = 64
  g1[2] = (int)(64u << 16);  // tensor_dim0 hi16=0    | tensor_dim1 lo16 = 64
  g1[3] = (int)(64u << 16);  // tensor_dim1 hi16=0    | tile_dim0 = 64
  g1[4] = 64;                // tile_dim1 = 64, tile_dim2 = 0
  g1[5] = 64;                // tensor_dim0_stride lo32 = 64
  g1[6] = 0;                 // stride0 hi16 = 0, tensor_dim1_stride lo16 = 0
  g1[7] = 0;
  i32x4 z4 = {0, 0, 0, 0};
#if __clang_major__ >= 23
  i32x8 z8 = {0, 0, 0, 0, 0, 0, 0, 0};
  __builtin_amdgcn_tensor_load_to_lds(g0, g1, z4, z4, z8, 0);
#else
  __builtin_amdgcn_tensor_load_to_lds(g0, g1, z4, z4, 0);
#endif
}
#else
#define HAVE_TDM 0
#endif

// ---------------------------------------------------------------------------
// x[n, :] = embed[x_idx[n], :]
__global__ void k_embed(const int* __restrict__ xidx,
                        const float* __restrict__ embed,
                        float* __restrict__ x) {
  int tid = blockIdx.x * blockDim.x + threadIdx.x;
  int n  = tid >> 5;
  int c4 = (tid & 31) * 4;
  float4 v = *reinterpret_cast<const float4*>(embed + (size_t)xidx[n] * CDIM + c4);
  *reinterpret_cast<float4*>(x + (size_t)n * CDIM + c4) = v;
}

// ---------------------------------------------------------------------------
// h = x.reshape(2N,64) @ W + bW  via V_WMMA_F32_16X16X4_F32 (fp32-exact).
// Block: 320 threads = 10 waves; each wave owns one 16-row tile, all 64 cols.
// Grid: 625 blocks => 6250 tiles exactly (no tail, EXEC always all-ones).
// W staged into LDS by the Tensor Data Mover (async DMA, TENSORcnt).
__global__ void __launch_bounds__(320) k_gemm_h(const float* __restrict__ x,
                                                const float* __restrict__ W,
                                                const float* __restrict__ bW,
                                                float* __restrict__ h) {
  __shared__ float4 LW4[64 * 64 / 4];           // 16 KB
  float* LW = reinterpret_cast<float*>(LW4);
  int tid = threadIdx.x;
#if HAVE_TDM
  if (tid < 32) {                                // wave 0 drives the DMA
    tdm_load_w64x64(W, LW);
    __builtin_amdgcn_s_wait_tensorcnt(0);
  }
#else
  for (int i = tid; i < 64 * 64 / 4; i += 320)
    LW4[i] = reinterpret_cast<const float4*>(W)[i];
#endif
  __syncthreads();

  int wave = tid >> 5;
  int lane = tid & 31;
  int tile = blockIdx.x * 10 + wave;            // < 6250 always
  int r0   = tile * 16;
  int lj   = lane & 15;
  int kh   = lane >> 4;

  // C/D layout: VGPR v, lanes 0-15 -> M=v, lanes 16-31 -> M=v+8; col = lj.
  v8f acc[4];
#pragma unroll
  for (int t = 0; t < 4; ++t) {
    float bv = bW[t * 16 + lj];
#pragma unroll
    for (int v = 0; v < 8; ++v) acc[t][v] = bv;
  }

  // A layout (16x4 f32): lane m (m<16) holds K={0,1}, lane m+16 holds K={2,3}
  const float* arow = x + (size_t)(r0 + lj) * 64 + 2 * kh;
#pragma unroll
  for (int s = 0; s < 16; ++s) {                // K = 4*s .. 4*s+3
    float2 av = *reinterpret_cast<const float2*>(arow + 4 * s);
    v2f a; a[0] = av.x; a[1] = av.y;
#pragma unroll
    for (int t = 0; t < 4; ++t) {
      // B layout (4x16 f32): vgpr0 lanes0-15 K=0, lanes16-31 K=2; vgpr1 K=1/K=3
      v2f b;
      b[0] = LW[(4 * s + 2 * kh)     * 64 + t * 16 + lj];
      b[1] = LW[(4 * s + 2 * kh + 1) * 64 + t * 16 + lj];
      acc[t] = __builtin_amdgcn_wmma_f32_16x16x4_f32(
          false, a, false, b, (short)0, acc[t], false, false);
    }
  }

#pragma unroll
  for (int t = 0; t < 4; ++t)
#pragma unroll
    for (int v = 0; v < 8; ++v)
      h[(size_t)(r0 + v + 8 * kh) * 64 + t * 16 + lj] = acc[t][v];
}

// ---------------------------------------------------------------------------
// sada[n, 0:4] = x[n] @ A_src ; sada[n, 4:8] = x[n] @ A_dst
__global__ void k_proj(const float* __restrict__ x,
                       const float* __restrict__ Asrc,
                       const float* __restrict__ Adst,
                       float* __restrict__ sada) {
  __shared__ float AA[CDIM * 8];
  int tid = threadIdx.x;
  for (int i = tid; i < CDIM * 4; i += blockDim.x) {
    int k = i >> 2, j = i & 3;
    AA[k * 8 + j]     = Asrc[i];
    AA[k * 8 + 4 + j] = Adst[i];
  }
  __syncthreads();
  int gid = blockIdx.x * blockDim.x + tid;
  if (gid >= NNODE * 8) return;
  int n = gid >> 3, j = gid & 7;
  const float* xr = x + (size_t)n * CDIM;
  float acc = 0.f;
#pragma unroll 8
  for (int k = 0; k < CDIM; ++k) acc = fmaf(xr[k], AA[k * 8 + j], acc);
  sada[gid] = acc;
}

// ---------------------------------------------------------------------------
// One wave32 per edge: lanes 0-3 compute the 2x2 tanh map, broadcast via shfl,
// every lane applies the map to a float2 slice of h[src] and scatters 4 atomics.
__global__ void __launch_bounds__(256) k_edge(const int* __restrict__ ei,
                                              const float* __restrict__ h,
                                              const float* __restrict__ sada,
                                              const float* __restrict__ bA,
                                              float* __restrict__ agg) {
  int wave = (blockIdx.x * blockDim.x + threadIdx.x) >> 5;  // == edge id, exact
  int lane = threadIdx.x & 31;
  int s = ei[wave];
  int d = ei[NEDGE + wave];
  float v = 0.f;
  if (lane < 4)
    v = tanhf(sada[s * 8 + lane] + sada[d * 8 + 4 + lane] + bA[lane]);
  float m00 = __shfl(v, 0, 32);   // maps[0][0]
  float m01 = __shfl(v, 1, 32);   // maps[0][1]
  float m10 = __shfl(v, 2, 32);   // maps[1][0]
  float m11 = __shfl(v, 3, 32);   // maps[1][1]
  const float* hs = h + (size_t)s * CDIM;
  float2 h0 = *reinterpret_cast<const float2*>(hs + 2 * lane);
  float2 h1 = *reinterpret_cast<const float2*>(hs + 64 + 2 * lane);
  float* ap = agg + (size_t)d * CDIM;
  atomic_add_f32(ap + 2 * lane,          m00 * h0.x + m01 * h1.x);
  atomic_add_f32(ap + 2 * lane + 1,      m00 * h0.y + m01 * h1.y);
  atomic_add_f32(ap + 64 + 2 * lane,     m10 * h0.x + m11 * h1.x);
  atomic_add_f32(ap + 64 + 2 * lane + 1, m10 * h0.y + m11 * h1.y);
}

// ---------------------------------------------------------------------------
__global__ void k_relu(const float* __restrict__ agg, float* __restrict__ x) {
  int i = blockIdx.x * blockDim.x + threadIdx.x;  // N*C/4 exact
  float4 v = reinterpret_cast<const float4*>(agg)[i];
  v.x = fmaxf(v.x, 0.f); v.y = fmaxf(v.y, 0.f);
  v.z = fmaxf(v.z, 0.f); v.w = fmaxf(v.w, 0.f);
  reinterpret_cast<float4*>(x)[i] = v;
}

// ---------------------------------------------------------------------------
// y[batch[n]] += relu(agg[n])  (final-layer relu fused into the pooling pass)
__global__ void k_pool(const float* __restrict__ agg, const int* __restrict__ batch,
                       float* __restrict__ y) {
  int tid = blockIdx.x * blockDim.x + threadIdx.x;  // N*32 exact
  int n = tid >> 5;
  int c = (tid & 31) * 4;
  int g = batch[n];
  float4 v = *reinterpret_cast<const float4*>(agg + (size_t)n * CDIM + c);
  float* yp = y + (size_t)g * CDIM + c;
  atomic_add_f32(yp,     fmaxf(v.x, 0.f));
  atomic_add_f32(yp + 1, fmaxf(v.y, 0.f));
  atomic_add_f32(yp + 2, fmaxf(v.z, 0.f));
  atomic_add_f32(yp + 3, fmaxf(v.w, 0.f));
}

// ---------------------------------------------------------------------------
// out[g] = relu(y[g] @ w1 + b1) @ w2 + b2   (one block of 64 threads per graph)
__global__ void __launch_bounds__(64) k_mlp(const float* __restrict__ y,
                                            const float* __restrict__ w1,
                                            const float* __restrict__ b1,
                                            const float* __restrict__ w2,
                                            const float* __restrict__ b2,
                                            float* __restrict__ out) {
  __shared__ float red[64];
  int g = blockIdx.x, j = threadIdx.x;
  const float* yr = y + (size_t)g * CDIM;
  float acc = b1[j];
#pragma unroll 8
  for (int k = 0; k < CDIM; ++k) acc = fmaf(yr[k], w1[k * HIDDEN + j], acc);
  red[j] = fmaxf(acc, 0.f) * w2[j];
  __syncthreads();
  for (int off = 32; off > 0; off >>= 1) {
    if (j < off) red[j] += red[j + off];
    __syncthreads();
  }
  if (j == 0) out[g] = red[0] + b2[0];
}

// ---------------------------------------------------------------------------
extern "C" void kernel_launch(void* const* d_in, const int* in_sizes, int n_in,
                              void* d_out, int out_size, void* d_ws, size_t ws_size,
                              hipStream_t stream) {
  const int*   x_idx = (const int*)d_in[0];
  const int*   ei    = (const int*)d_in[1];
  const int*   batch = (const int*)d_in[2];
  const float* embed = (const float*)d_in[3];
  const float* W     = (const float*)d_in[4];
  const float* bW    = (const float*)d_in[5];
  const float* A_src = (const float*)d_in[6];
  const float* A_dst = (const float*)d_in[7];
  const float* bA    = (const float*)d_in[8];
  const float* w1    = (const float*)d_in[9];
  const float* b1    = (const float*)d_in[10];
  const float* w2    = (const float*)d_in[11];
  const float* b2    = (const float*)d_in[12];
  float* out = (float*)d_out;

  char* ws = (char*)d_ws;
  size_t off = 0;
  auto alloc = [&](size_t bytes) {
    void* p = ws + off;
    off = (off + bytes + 255) & ~(size_t)255;
    return p;
  };
  float* x    = (float*)alloc((size_t)NNODE * CDIM * 4);
  float* h    = (float*)alloc((size_t)M2 * HIDDEN * 4);
  float* agg  = (float*)alloc((size_t)NNODE * CDIM * 4);
  float* sada = (float*)alloc((size_t)NNODE * 8 * 4);
  float* y    = (float*)alloc((size_t)NGRAPH * CDIM * 4);

  k_embed<<<NNODE * 32 / 256, 256, 0, stream>>>(x_idx, embed, x);

  for (int l = 0; l < NLAYER; ++l) {
    hipMemsetAsync(agg, 0, (size_t)NNODE * CDIM * 4, stream);
    k_gemm_h<<<M2 / 16 / 10, 320, 0, stream>>>(x, W + l * HIDDEN * HIDDEN,
                                               bW + l * HIDDEN, h);
    k_proj<<<(NNODE * 8 + 255) / 256, 256, 0, stream>>>(
        x, A_src + l * CDIM * 4, A_dst + l * CDIM * 4, sada);
    k_edge<<<NEDGE * 32 / 256, 256, 0, stream>>>(ei, h, sada, bA + l * 4, agg);
    if (l + 1 < NLAYER)  // final relu is fused into k_pool
      k_relu<<<NNODE * CDIM / 4 / 256, 256, 0, stream>>>(agg, x);
  }

  hipMemsetAsync(y, 0, (size_t)NGRAPH * CDIM * 4, stream);
  k_pool<<<NNODE * 32 / 256, 256, 0, stream>>>(agg, batch, y);
  k_mlp<<<NGRAPH, 64, 0, stream>>>(y, w1, b1, w2, b2, out);
}